// BayesianSTDPAdaptive_9182640079056
// MI455X (gfx1250) — compile-verified
//
#include <hip/hip_runtime.h>

typedef float v2f __attribute__((ext_vector_type(2)));
typedef float v8f __attribute__((ext_vector_type(8)));

#define TB      10
#define ITERS   200
#define I_SZ    1024
#define O_SZ    512

// ---------------------------------------------------------------------------
// Kernel 1: tos[k,o] = sum_b spk[k,b,o] ; tsum[k] = sum_o tos[k,o]
// grid = ITERS blocks, O_SZ threads
// ---------------------------------------------------------------------------
__global__ void __launch_bounds__(O_SZ)
stdp_reduce_kernel(const float* __restrict__ spk,
                   float* __restrict__ tos,
                   float* __restrict__ tsum) {
    const int k = blockIdx.x;
    const int o = threadIdx.x;
    const float* base = spk + k * TB * O_SZ + o;
    float s = 0.0f;
#pragma unroll
    for (int b = 0; b < TB; ++b) s += base[b * O_SZ];
    tos[k * O_SZ + o] = s;

    __shared__ float red[O_SZ];
    red[o] = s;
    __syncthreads();
    for (int off = O_SZ / 2; off > 0; off >>= 1) {
        if (o < off) red[o] += red[o + off];
        __syncthreads();
    }
    if (o == 0) tsum[k] = red[0];
}

// ---------------------------------------------------------------------------
// Kernel 2: 200-step serial bias recurrence, one thread per output neuron
// ---------------------------------------------------------------------------
__global__ void __launch_bounds__(O_SZ)
stdp_bias_kernel(const float* __restrict__ biases,
                 const float* __restrict__ mu_bias,
                 const float* __restrict__ tos,
                 const float* __restrict__ tsum,
                 float* __restrict__ b_out) {
    const int o = threadIdx.x;
    float b  = biases[o];
    float mu = mu_bias[o];
    for (int k = 0; k < ITERS; ++k) {
        float t  = tos[k * O_SZ + o];
        float ts = tsum[k];
        b += mu * ((__expf(-b) * t - 1.0f) * ts);
    }
    b_out[o] = b;
}

// ---------------------------------------------------------------------------
// Kernel 3: fused corr-GEMM (V_WMMA_F32_16X16X4_F32, K=10 as 4+4+2pad) + scan.
// One wave32 owns one 16x16 tile of W for all 200 iterations.
// C/D layout: lane L -> N = L&15 ; element j -> M = j + 8*(L>>4).
// A (16x4 f32): lanes 0-15 hold M, VGPR0={K0|K2}, VGPR1={K1|K3} (lo/hi halves)
// B (4x16 f32): symmetric with N in lanes.
// grid: 256 blocks x 256 threads = 2048 waves = (512/16)*(1024/16) tiles.
// ---------------------------------------------------------------------------
__global__ void __launch_bounds__(256)
stdp_weight_kernel(const float* __restrict__ psp,
                   const float* __restrict__ spk,
                   const float* __restrict__ weights,
                   const float* __restrict__ mu_w,
                   const float* __restrict__ tos,
                   float* __restrict__ w_out) {
    const int wave = (blockIdx.x * blockDim.x + threadIdx.x) >> 5;
    const int lane = threadIdx.x & 31;
    const int to = wave >> 6;          // 0..31   O-tile
    const int ti = wave & 63;          // 0..63   I-tile
    const int o0 = to * 16;
    const int i0 = ti * 16;
    const int half = lane >> 4;        // 0 or 1
    const int col  = lane & 15;        // N (and M for A-loads)
    const int rowBase = o0 + 8 * half; // M rows this lane's elements live on

    // Resident tile of w and mu (8 elements per lane, WMMA C layout).
    float w[8], mu[8];
#pragma unroll
    for (int j = 0; j < 8; ++j) {
        int idx = (rowBase + j) * I_SZ + i0 + col;
        w[j]  = weights[idx];
        mu[j] = mu_w[idx];
    }

    for (int k = 0; k < ITERS; ++k) {
        const int tbase = k * TB;      // global time index of this chunk

        v8f c = {};                    // zero accumulator
#pragma unroll
        for (int kb = 0; kb < 12; kb += 4) {
            // This lane supplies K rows (kb + 2*half) and (kb + 2*half + 1).
            int b0 = kb + 2 * half;
            // Mask rows >= TB by multiplying with 0 (no EXEC divergence).
            float m0 = (b0     < TB) ? 1.0f : 0.0f;
            float m1 = (b0 + 1 < TB) ? 1.0f : 0.0f;
            int r0 = (b0     < TB) ? b0     : (TB - 1);
            int r1 = (b0 + 1 < TB) ? b0 + 1 : (TB - 1);

            v2f a, bm;
            a.x  = spk[(tbase + r0) * O_SZ + o0 + col] * m0;
            a.y  = spk[(tbase + r1) * O_SZ + o0 + col] * m1;
            bm.x = psp[(tbase + r0) * I_SZ + i0 + col] * m0;
            bm.y = psp[(tbase + r1) * I_SZ + i0 + col] * m1;

            c = __builtin_amdgcn_wmma_f32_16x16x4_f32(
                    /*neg_a=*/false, a, /*neg_b=*/false, bm,
                    /*c_mod=*/(short)0, c,
                    /*reuse_a=*/false, /*reuse_b=*/false);
        }

        // Elementwise scan step: w += mu * (corr * exp(-w) - tos[o])
#pragma unroll
        for (int j = 0; j < 8; ++j) {
            float t = tos[k * O_SZ + rowBase + j];
            float e = __expf(-w[j]);
            w[j] += mu[j] * (c[j] * e - t);
        }
    }

#pragma unroll
    for (int j = 0; j < 8; ++j)
        w_out[(rowBase + j) * I_SZ + i0 + col] = w[j];
}

// ---------------------------------------------------------------------------
extern "C" void kernel_launch(void* const* d_in, const int* in_sizes, int n_in,
                              void* d_out, int out_size, void* d_ws, size_t ws_size,
                              hipStream_t stream) {
    const float* psp     = (const float*)d_in[0];  // (2000, 1024)
    const float* spk     = (const float*)d_in[1];  // (2000, 512)
    const float* weights = (const float*)d_in[2];  // (512, 1024)
    const float* biases  = (const float*)d_in[3];  // (512,)
    const float* mu_w    = (const float*)d_in[4];  // (512, 1024)
    const float* mu_b    = (const float*)d_in[5];  // (512,)

    float* out  = (float*)d_out;                   // w (512*1024) then b (512)
    float* tos  = (float*)d_ws;                    // ITERS*O_SZ floats
    float* tsum = tos + ITERS * O_SZ;              // ITERS floats

    stdp_reduce_kernel<<<ITERS, O_SZ, 0, stream>>>(spk, tos, tsum);
    stdp_bias_kernel<<<1, O_SZ, 0, stream>>>(biases, mu_b, tos, tsum,
                                             out + O_SZ * I_SZ);
    stdp_weight_kernel<<<256, 256, 0, stream>>>(psp, spk, weights, mu_w, tos, out);
}